// DictLearn_32109175505530
// MI455X (gfx1250) — compile-verified
//
#include <hip/hip_runtime.h>
#include <math.h>

#define NEMB 512      // K: number of atoms
#define EDIM 64       // D: embedding dim
#define NCOL 32768    // N: columns of zf
#define TOTAL (EDIM * NCOL)   // 2097152

// d_out layout (floats): loss | recon_out | zf | perplexity | gamma
#define OFF_LOSS  0
#define OFF_RECON 1
#define OFF_ZF    (1 + TOTAL)
#define OFF_PERP  (1 + 2 * TOTAL)
#define OFF_GAMMA (2 + 2 * TOTAL)

// workspace layout (floats)
#define WS_ACC 0                      // [0]=sum sq err, [1]=sum entropy
#define WS_DN  16                     // Dn [64][512]
#define WS_G   (16 + EDIM * NEMB)     // G  [512][512]

typedef float v2f __attribute__((ext_vector_type(2)));
typedef float v8f __attribute__((ext_vector_type(8)));

// ---------------------------------------------------------------- prep
// zf_flat[i] = z_e[b,c,h,w] with i = ((b*32+h)*32+w)*64+c ; zero accumulators
__global__ void prep_kernel(const float* __restrict__ z_e,
                            float* __restrict__ out,
                            float* __restrict__ ws) {
  int i = blockIdx.x * blockDim.x + threadIdx.x;
  if (i < 8) ws[WS_ACC + i] = 0.0f;
  if (i < TOTAL) {
    int c = i & 63, w = (i >> 6) & 31, h = (i >> 11) & 31, b = i >> 16;
    out[OFF_ZF + i] = z_e[((b * 64 + c) * 32 + h) * 32 + w];
  }
}

// ---------------------------------------------------------------- norm
__global__ void norm_kernel(const float* __restrict__ dict, float* __restrict__ ws) {
  __shared__ float red[64];
  int k = blockIdx.x, d = threadIdx.x;
  float v = dict[d * NEMB + k];
  red[d] = v * v;
  __syncthreads();
  for (int s = 32; s > 0; s >>= 1) {
    if (d < s) red[d] += red[d + s];
    __syncthreads();
  }
  ws[WS_DN + d * NEMB + k] = v / sqrtf(red[0]);
}

// ---------------------------------------------------------------- gram (WMMA)
// G = Dn^T Dn. One wave per 16x16 tile, f32 WMMA 16x16x4, 16 k-steps over D=64.
__global__ void gram_kernel(const float* __restrict__ dn, float* __restrict__ g) {
  int tm = blockIdx.x >> 5, tn = blockIdx.x & 31;
  int lane = threadIdx.x & 31;
  int half = lane >> 4, l16 = lane & 15;
  v8f acc = {};
  for (int kk = 0; kk < 16; ++kk) {
    int d0 = kk * 4 + 2 * half;   // A/B vgpr i, half h -> K = 2h + i
    v2f a, b;
    a[0] = dn[(d0    ) * NEMB + tm * 16 + l16];
    a[1] = dn[(d0 + 1) * NEMB + tm * 16 + l16];
    b[0] = dn[(d0    ) * NEMB + tn * 16 + l16];
    b[1] = dn[(d0 + 1) * NEMB + tn * 16 + l16];
    acc = __builtin_amdgcn_wmma_f32_16x16x4_f32(false, a, false, b,
                                                (short)0, acc, false, false);
  }
  int rbase = tm * 16 + half * 8;           // C/D: vgpr v -> M = v + 8*half
  for (int v = 0; v < 8; ++v)
    g[(rbase + v) * NEMB + tn * 16 + l16] = acc[v];
}

// ---------------------------------------------------------------- fused OMP
// One block = 16 columns. 8 waves build h_bar[512x16] in LDS via WMMA
// (each wave: 4 row-tiles x 16 k-steps). Then 16 threads run the serial
// Batch-OMP recursion; the rank-k h update runs on all 256 threads.
__global__ __launch_bounds__(256) void omp_kernel(float* __restrict__ out,
                                                  float* __restrict__ ws) {
  const float* Dn = ws + WS_DN;
  const float* G  = ws + WS_G;
  const float* zf = out + OFF_ZF;

  __shared__ float Hs[NEMB * 16];     // current h (later reused as gamma block)
  __shared__ float coefL[16][8];
  __shared__ float xL[16][8];
  __shared__ int   idxL[16][8];

  int tid  = threadIdx.x;
  int lane = tid & 31, wave = tid >> 5;
  int half = lane >> 4, l16 = lane & 15;
  int n0 = blockIdx.x * 16;

  // --- h_bar = Dn^T * zf_block via WMMA f32 16x16x4 ---
  for (int tt = 0; tt < 4; ++tt) {
    int a0 = (wave * 4 + tt) * 16;    // atom base for this 16-row tile
    v8f acc = {};
    for (int kk = 0; kk < 16; ++kk) {
      int d0 = kk * 4 + 2 * half;
      v2f a, b;
      a[0] = Dn[(d0    ) * NEMB + a0 + l16];
      a[1] = Dn[(d0 + 1) * NEMB + a0 + l16];
      b[0] = zf[(size_t)(d0    ) * NCOL + n0 + l16];
      b[1] = zf[(size_t)(d0 + 1) * NCOL + n0 + l16];
      acc = __builtin_amdgcn_wmma_f32_16x16x4_f32(false, a, false, b,
                                                  (short)0, acc, false, false);
    }
    for (int v = 0; v < 8; ++v)
      Hs[(a0 + half * 8 + v) * 16 + l16] = acc[v];
  }
  __syncthreads();

  // --- serial OMP state (only threads 0..15 use it; column c = tid) ---
  float Lm[64];        // lower-triangular 8x8, row-major
  float hb[8];         // h_bar at selected indices
  float xo[8];         // previous-step solution
  float xn[8];         // current solution
  int   idxs[8];
  unsigned m[16];
  if (tid < 16) for (int i = 0; i < 16; ++i) m[i] = 0u;

  for (int s = 1; s <= 8; ++s) {
    if (tid < 16) {
      int c = tid;
      // argmax |h| over unmasked atoms
      float best = -1.0f; int bidx = 0;
      for (int r = 0; r < NEMB; ++r) {
        if (!((m[r >> 5] >> (r & 31)) & 1u)) {
          float av = fabsf(Hs[r * 16 + c]);
          if (av > best) { best = av; bidx = r; }
        }
      }
      m[bidx >> 5] |= 1u << (bidx & 31);
      // progressive Cholesky update of L
      if (s == 1) {
        Lm[0] = 1.0f;
      } else {
        int sz = s - 1;
        float w[8];
        for (int i = 0; i < sz; ++i) {            // solve L w = g
          float a = G[(size_t)idxs[i] * NEMB + bidx];
          for (int j = 0; j < i; ++j) a -= Lm[i * 8 + j] * w[j];
          w[i] = a / Lm[i * 8 + i];
        }
        float ssum = 0.0f;
        for (int i = 0; i < sz; ++i) { Lm[sz * 8 + i] = w[i]; ssum += w[i] * w[i]; }
        Lm[sz * 8 + sz] = sqrtf(fmaxf(1.0f - ssum, 1e-20f));
      }
      idxs[s - 1] = bidx;
      idxL[c][s - 1] = bidx;
      // reconstruct h_bar[bidx] from current h:  hbar = h + sum xo_j * G[idx_j, bidx]
      float hbn = Hs[bidx * 16 + c];
      for (int j = 0; j < s - 1; ++j) hbn += xo[j] * G[(size_t)idxs[j] * NEMB + bidx];
      hb[s - 1] = hbn;
      // solve L y = hb ; L^T xn = y   (size s)
      float y[8];
      for (int i = 0; i < s; ++i) {
        float a = hb[i];
        for (int j = 0; j < i; ++j) a -= Lm[i * 8 + j] * y[j];
        y[i] = a / Lm[i * 8 + i];
      }
      for (int i = s - 1; i >= 0; --i) {
        float a = y[i];
        for (int j = i + 1; j < s; ++j) a -= Lm[j * 8 + i] * xn[j];
        xn[i] = a / Lm[i * 8 + i];
      }
      // delta coefficients: h_new = h_old + sum coef_j * G[idx_j, :]
      for (int j = 0; j < s - 1; ++j) coefL[c][j] = xo[j] - xn[j];
      coefL[c][s - 1] = -xn[s - 1];
      for (int j = 0; j < s; ++j) xo[j] = xn[j];
      if (s == 8) for (int j = 0; j < 8; ++j) xL[c][j] = xn[j];
    }
    __syncthreads();
    if (s < 8) {   // parallel rank-s update of h across all 256 threads
      for (int e = tid; e < NEMB * 16; e += 256) {
        int r = e >> 4, c = e & 15;
        float a = Hs[e];
        for (int j = 0; j < s; ++j)
          a += coefL[c][j] * G[(size_t)idxL[c][j] * NEMB + r];
        Hs[e] = a;
      }
    }
    __syncthreads();
  }

  // --- gamma block: scatter sparse x into reused Hs, stream out ---
  for (int e = tid; e < NEMB * 16; e += 256) Hs[e] = 0.0f;
  __syncthreads();
  if (tid < 16) for (int j = 0; j < 8; ++j) Hs[idxs[j] * 16 + tid] = xn[j];
  __syncthreads();
  float* gam = out + OFF_GAMMA;
  for (int e = tid; e < NEMB * 16; e += 256) {
    int r = e >> 4, c = e & 15;
    gam[(size_t)r * NCOL + n0 + c] = Hs[e];
  }

  // --- recon (permuted to [b,c,h,w]) + squared-error partial ---
  float lacc = 0.0f;
  for (int e = tid; e < EDIM * 16; e += 256) {
    int d = e >> 4, c = e & 15;
    float rsum = 0.0f;
    for (int j = 0; j < 8; ++j)
      rsum += xL[c][j] * Dn[(size_t)d * NEMB + idxL[c][j]];
    int fi = d * NCOL + n0 + c;                 // flat index in [64,32768]
    int ch = fi & 63, wq = (fi >> 6) & 31, hq = (fi >> 11) & 31, bq = fi >> 16;
    out[OFF_RECON + ((bq * 64 + ch) * 32 + hq) * 32 + wq] = rsum;
    float diff = rsum - zf[fi];
    lacc += diff * diff;
  }
  atomicAdd(ws + WS_ACC + 0, lacc);

  // --- per-column softmax entropy over 512 entries (504 zeros + 8 vals) ---
  if (tid < 16) {
    float gmax = 0.0f;
    for (int j = 0; j < 8; ++j) gmax = fmaxf(gmax, xn[j]);
    float z0 = expf(-gmax);
    float Z = 504.0f * z0;
    float ex[8];
    for (int j = 0; j < 8; ++j) { ex[j] = expf(xn[j] - gmax); Z += ex[j]; }
    float p0 = z0 / Z;
    float S = 504.0f * p0 * logf(p0 + 1e-10f);
    for (int j = 0; j < 8; ++j) { float p = ex[j] / Z; S += p * logf(p + 1e-10f); }
    atomicAdd(ws + WS_ACC + 1, S);
  }
}

// ---------------------------------------------------------------- finalize
__global__ void finalize_kernel(const float* __restrict__ beta,
                                const float* __restrict__ ws,
                                float* __restrict__ out) {
  if (threadIdx.x == 0 && blockIdx.x == 0) {
    float mse = ws[WS_ACC + 0] / (float)TOTAL;
    out[OFF_LOSS] = mse * 0.25f + beta[0] * mse;   // e_latent*0.25 + beta0*q
    out[OFF_PERP] = expf(-(ws[WS_ACC + 1] / (float)NCOL));
  }
}

extern "C" void kernel_launch(void* const* d_in, const int* in_sizes, int n_in,
                              void* d_out, int out_size, void* d_ws, size_t ws_size,
                              hipStream_t stream) {
  (void)in_sizes; (void)n_in; (void)out_size; (void)ws_size;
  const float* z_e  = (const float*)d_in[0];
  const float* dict = (const float*)d_in[1];
  const float* beta = (const float*)d_in[2];
  float* out = (float*)d_out;
  float* ws  = (float*)d_ws;

  prep_kernel<<<TOTAL / 256, 256, 0, stream>>>(z_e, out, ws);
  norm_kernel<<<NEMB, 64, 0, stream>>>(dict, ws);
  gram_kernel<<<(NEMB / 16) * (NEMB / 16), 32, 0, stream>>>(ws + WS_DN, ws + WS_G);
  omp_kernel<<<NCOL / 16, 256, 0, stream>>>(out, ws);
  finalize_kernel<<<1, 32, 0, stream>>>(beta, ws, out);
}